// UnfiltBackproj3DSeparable1D2DForLoop_6416681140698
// MI455X (gfx1250) — compile-verified
//
#include <hip/hip_runtime.h>
#include <hip/hip_bf16.h>
#include <stdint.h>

// Out[M=8192, N=4096] = Y[M,K=64] x W12[K,N],  W12[k,n] = W1[k, n>>4] * W2[k, n&15]
//   m = ((b*4 + x)*8 + yy)*8 + z   -> y flat offset(m,k) = (m>>8)*16384 + k*256 + (m&255)
//   n = (i*4 + j)*4 + l            -> out offset(m,n)   = (m>>8)*1048576 + x*262144
//                                      + (n>>4)*1024 + yy*128 + ((n>>2)&3)*32 + z*4 + (n&3)
// Store-bound: 128 MiB fp32 streamed out (NT stores) vs 4.3 GFLOP via f32 WMMA.

typedef __attribute__((ext_vector_type(2))) float v2f;
typedef __attribute__((ext_vector_type(8))) float v8f;
typedef __attribute__((ext_vector_type(4))) int   v4i;

#if defined(__AMDGCN__) && __has_builtin(__builtin_amdgcn_global_load_async_to_lds_b128)
#define HAVE_ASYNC_LDS 1
typedef __attribute__((address_space(1))) v4i as1_v4i;   // global int4 (prints as __device__*)
typedef __attribute__((address_space(3))) v4i as3_v4i;   // LDS int4
#endif

__global__ __launch_bounds__(256)
void backproj_wmma_kernel(const float* __restrict__ Y,
                          const float* __restrict__ W1,   // [64][256]
                          const float* __restrict__ W2,   // [64][16]
                          float* __restrict__ Out)
{
    // A tile staged as [k][m]: 64 lines x 16 floats (64B) -> 4 KB, 16B-chunk friendly
    __shared__ float Alds[64 * 16];

    const int t      = threadIdx.x;
    const int m_base = blockIdx.x * 16;   // 512 m-tiles
    const int wgn    = blockIdx.y;        // 32 n-tiles of 128
    const int gbase  = (m_base >> 8) * 16384 + (m_base & 255);

    // ---- Stage A tile: each of 256 threads moves one 16B chunk ----
    {
        const int k = t >> 2;             // 0..63
        const int q = (t & 3) << 2;       // m offset 0,4,8,12
        const float* src = Y + gbase + k * 256 + q;
        const int lofs = k * 16 + q;
#ifdef HAVE_ASYNC_LDS
        // CDNA5 async copy: ASYNCcnt-tracked, no VGPR data staging
        as1_v4i* gsrc = (as1_v4i*)(uintptr_t)src;
        as3_v4i* ldst = (as3_v4i*)(uint32_t)(uintptr_t)&Alds[lofs];
        __builtin_amdgcn_global_load_async_to_lds_b128(gsrc, ldst, 0, 0);
#else
        const float4 v = *(const float4*)src;
        *(float4*)&Alds[lofs] = v;
#endif
    }
#ifdef HAVE_ASYNC_LDS
#if __has_builtin(__builtin_amdgcn_s_wait_asynccnt)
    __builtin_amdgcn_s_wait_asynccnt(0);
#else
    asm volatile("s_wait_asynccnt 0x0" ::: "memory");
#endif
#endif
    __syncthreads();

    const int wave   = t >> 5;
    const int lane   = t & 31;
    const int nl     = lane & 15;         // N within tile / A m-row
    const int g      = lane >> 4;         // lane group: K pair select
    const int n_base = wgn * 128 + wave * 16;
    const int i_col  = n_base >> 4;       // wave-uniform W1 column

    v8f acc = {};

    // ---- K=64 as 16 chained 16x16x4 f32 WMMAs ----
    // A lane layout: (m = lane&15, k = 4s + 2g + v);  LDS [k][m] -> 2x b32 (2addr-fusable)
    // B lane layout (mirrors A): (k = 4s + 2g + v, n = lane&15)
    #pragma unroll
    for (int s = 0; s < 16; ++s) {
        const int k0 = 4 * s + 2 * g;
        v2f a;
        a.x = Alds[k0 * 16 + nl];
        a.y = Alds[(k0 + 1) * 16 + nl];
        v2f b;
        b.x = W1[k0 * 256 + i_col]       * W2[k0 * 16 + nl];
        b.y = W1[(k0 + 1) * 256 + i_col] * W2[(k0 + 1) * 16 + nl];
        acc = __builtin_amdgcn_wmma_f32_16x16x4_f32(
            /*neg_a=*/false, a, /*neg_b=*/false, b,
            /*c_mod=*/(short)0, acc, /*reuse_a=*/false, /*reuse_b=*/false);
    }

    // ---- Scatter-store D: VGPR r -> (m = m_base + r + 8g, n = n_base + nl) ----
    // The 4 lanes sharing j and g (l = 0..3), over the 8 unrolled stores, fill a contiguous 128B output region.
    const int j = nl >> 2, l = nl & 3;
    const int ncomp = i_col * 1024 + j * 32 + l;
    const int bofs  = (m_base >> 8) * 1048576;   // batch component (tile never crosses b)
    #pragma unroll
    for (int r = 0; r < 8; ++r) {
        const int m   = m_base + r + 8 * g;
        const int rem = m & 255;                  // x*64 + yy*8 + z
        const int off = bofs + (rem >> 6) * 262144 + ((rem >> 3) & 7) * 128
                      + (rem & 7) * 4 + ncomp;
        __builtin_nontemporal_store(acc[r], Out + off);  // streaming, no reuse -> TH_NT
    }
}

extern "C" void kernel_launch(void* const* d_in, const int* in_sizes, int n_in,
                              void* d_out, int out_size, void* d_ws, size_t ws_size,
                              hipStream_t stream) {
    const float* y  = (const float*)d_in[0];   // (32,64,4,8,8)
    const float* w1 = (const float*)d_in[1];   // (64,1,256,1,1) -> [64][256]
    const float* w2 = (const float*)d_in[2];   // (64,1,1,4,4)   -> [64][16]
    float* out = (float*)d_out;                // (32,1,1024,32,32) fp32

    dim3 grid(512, 32);   // 512 m-tiles x 32 n-tiles(128)
    backproj_wmma_kernel<<<grid, dim3(256), 0, stream>>>(y, w1, w2, out);
}